// TemporalAbstraction_Attention_4398046511724
// MI455X (gfx1250) — compile-verified
//
#include <hip/hip_runtime.h>
#include <hip/hip_bf16.h>

// ---------------------------------------------------------------------------
// Flow-embedding layer for MI455X (gfx1250, wave32).
//   KNN -> gather/concat -> 3x (1x1 conv as WMMA f32 GEMM + train-mode BN + ReLU)
//   -> max over k.
// GEMMs use V_WMMA_F32_16X16X4_F32. Weights are staged in LDS (padded pitch to
// avoid bank conflicts); each wave owns TWO 16-sample B fragments so every
// A fragment (ds_load_b64) feeds two WMMAs. Activations are channel-major
// [C][S] (S = B*N*K = 524288) so B loads, BN reductions and the max-pool are
// coalesced; BN+ReLU is fused into the next layer's operand load.
// ---------------------------------------------------------------------------

#define B_     8
#define N_     4096
#define C_     64
#define K_     16
#define S_     (B_ * N_ * K_)        // 524288 samples
#define INCH0  131
#define INCH0P 132                   // padded to multiple of 4 for k-loop
#define EPS_   1e-5f

typedef __attribute__((ext_vector_type(2))) float v2f;
typedef __attribute__((ext_vector_type(8))) float v8f;

// ----- workspace layout (bytes) -----
#define OFF_IDX   0ull                                   // B*N*K ints  = 2 MiB
#define OFF_W0P   (2ull << 20)                           // 64*132 f32
#define OFF_STATS ((2ull << 20) + (64ull << 10))         // 6*128 f32
#define OFF_X0    (4ull << 20)                           // [132][S] f32 = 277 MB
#define SZ_X0     ((unsigned long long)INCH0P * S_ * 4ull)
#define OFF_Y1    (OFF_X0 + SZ_X0)                       // [64][S] = 134 MB
#define OFF_Y2    (OFF_Y1 + 64ull * S_ * 4ull)           // [64][S] = 134 MB
#define OFF_Y3    OFF_X0                                 // [128][S] aliases x0

// ---------------------------------------------------------------------------
// Pad W0 [64][131] -> [64][132] with a zero column (x0 row 131 is zeroed too).
__global__ void pad_w0_kernel(const float* __restrict__ W0, float* __restrict__ W0p) {
    int t = blockIdx.x * 256 + threadIdx.x;
    if (t >= 64 * INCH0P) return;
    int o = t / INCH0P, c = t % INCH0P;
    W0p[t] = (c < INCH0) ? W0[o * INCH0 + c] : 0.0f;
}

// ---------------------------------------------------------------------------
// Brute-force KNN: one thread per query point, refs tiled through LDS,
// top-16 kept sorted in registers (insert + bubble, fully unrolled).
__global__ void knn_kernel(const float* __restrict__ q,   // pos1_re [B][3][N]
                           const float* __restrict__ r,   // pos2    [B][3][N]
                           int* __restrict__ idx) {
    __shared__ float sx[256], sy[256], sz[256];
    int b = blockIdx.x >> 4;                       // 16 blocks per batch
    int n = ((blockIdx.x & 15) << 8) + threadIdx.x;
    const float* qb = q + b * 3 * N_;
    float qx = qb[0 * N_ + n], qy = qb[1 * N_ + n], qz = qb[2 * N_ + n];

    float bestd[K_];
    int   besti[K_];
#pragma unroll
    for (int t = 0; t < K_; ++t) { bestd[t] = 3.4e38f; besti[t] = 0; }

    const float* rb = r + b * 3 * N_;
    for (int tile = 0; tile < N_; tile += 256) {
        __syncthreads();
        sx[threadIdx.x] = rb[0 * N_ + tile + threadIdx.x];
        sy[threadIdx.x] = rb[1 * N_ + tile + threadIdx.x];
        sz[threadIdx.x] = rb[2 * N_ + tile + threadIdx.x];
        __syncthreads();
        for (int j = 0; j < 256; ++j) {
            float dx = qx - sx[j], dy = qy - sy[j], dz = qz - sz[j];
            float d = dx * dx + dy * dy + dz * dz;
            if (d < bestd[K_ - 1]) {
                bestd[K_ - 1] = d; besti[K_ - 1] = tile + j;
#pragma unroll
                for (int t = K_ - 1; t >= 1; --t) {
                    if (bestd[t] < bestd[t - 1]) {
                        float td = bestd[t]; bestd[t] = bestd[t - 1]; bestd[t - 1] = td;
                        int   ti = besti[t]; besti[t] = besti[t - 1]; besti[t - 1] = ti;
                    }
                }
            }
        }
    }
    int base = (b * N_ + n) * K_;
#pragma unroll
    for (int t = 0; t < K_; ++t) idx[base + t] = besti[t];
}

// ---------------------------------------------------------------------------
// Build x0 [132][S], channel-major: rows 0..2 pos_diff, 3..66 gathered feat2,
// 67..130 repeated feat1, row 131 zero pad. One thread per sample; for a fixed
// channel, consecutive threads write consecutive s -> coalesced stores.
__global__ void build_x0_kernel(const float* __restrict__ pos1,
                                const float* __restrict__ pos2,
                                const float* __restrict__ feat1,
                                const float* __restrict__ feat2,
                                const int* __restrict__ idx,
                                float* __restrict__ x0) {
    int s = blockIdx.x * 256 + threadIdx.x;       // s = ((b*N + n)*K + k)
    int n = (s >> 4) & (N_ - 1);
    int b = s >> 16;
    int j = idx[s];
    const float* p1 = pos1 + b * 3 * N_;
    const float* p2 = pos2 + b * 3 * N_;
#pragma unroll
    for (int d = 0; d < 3; ++d)
        x0[d * S_ + s] = p2[d * N_ + j] - p1[d * N_ + n];
    const float* f2 = feat2 + b * C_ * N_;
#pragma unroll 4
    for (int c = 0; c < C_; ++c)
        x0[(3 + c) * S_ + s] = f2[c * N_ + j];
    const float* f1 = feat1 + b * C_ * N_;
#pragma unroll 4
    for (int c = 0; c < C_; ++c)
        x0[(67 + c) * S_ + s] = f1[c * N_ + n];
    x0[INCH0 * S_ + s] = 0.0f;
}

// ---------------------------------------------------------------------------
// WMMA GEMM: Y[out][S] = W[out][in] * f(X[in][S]),  f = BN+ReLU or identity.
// Weights staged in LDS with pitch in_ch+2 (breaks bank conflicts: 66 mod 64
// = 2, 134 mod 64 = 6, both give conflict-free 16-lane row access).
// Each wave owns 32 samples = two 16-wide B fragments; every A fragment
// (ds_load_b64) feeds two WMMAs. Block = 8 waves = 256 samples.
// Lane mapping (V_WMMA_F32_16X16X4_F32):
//   A 16x4: lane L -> v2f { W[L%16][kk+2*(L/16)], W[L%16][kk+2*(L/16)+1] }
//   B 4x16: lane L -> v2f { X[kk+2*(L/16)][s0+L%16], X[kk+2*(L/16)+1][s0+L%16] }
//   D 16x16: lane L, reg r -> Y[t*16 + r + 8*(L/16)][s0 + L%16]
template <int OUT_TILES>
__global__ void gemm_wmma_kernel(const float* __restrict__ W, int in_ch,
                                 const float* __restrict__ X,
                                 const float* __restrict__ scale,
                                 const float* __restrict__ shift,
                                 int apply_bn,
                                 float* __restrict__ Y) {
    __shared__ float sW[8576];                     // max(64*134, 128*66) floats
    const int pitch  = in_ch + 2;
    const int out_ch = OUT_TILES * 16;

    // cooperative stage: W[out][in] -> sW[out][pitch]
    for (int i = threadIdx.x; i < out_ch * in_ch; i += 256) {
        int o = i / in_ch, c = i - o * in_ch;
        sW[o * pitch + c] = W[i];
    }
    __syncthreads();

    const int wave = threadIdx.x >> 5;
    const int lane = threadIdx.x & 31;
    const int col  = lane & 15;
    const int half = lane >> 4;
    const int s0   = (blockIdx.x * 8 + wave) * 32; // 32 samples per wave

    v8f acc[OUT_TILES][2];
#pragma unroll
    for (int t = 0; t < OUT_TILES; ++t) {
        acc[t][0] = (v8f){};
        acc[t][1] = (v8f){};
    }

    // B-operand pointers for the two fragments; channel row starts at 2*half.
    const float* xA = X + (2 * half) * S_ + s0 + col;
    const float* xB = xA + 16;
    int aoff = col * pitch + 2 * half;             // LDS A offset (floats)

    for (int kk = 0; kk < in_ch; kk += 4) {
        const int c0 = kk + 2 * half;
        float b0A = xA[0], b1A = xA[S_];
        float b0B = xB[0], b1B = xB[S_];
        if (apply_bn) {
            float sc0 = scale[c0], sh0 = shift[c0];
            float sc1 = scale[c0 + 1], sh1 = shift[c0 + 1];
            b0A = fmaxf(0.0f, fmaf(sc0, b0A, sh0));
            b1A = fmaxf(0.0f, fmaf(sc1, b1A, sh1));
            b0B = fmaxf(0.0f, fmaf(sc0, b0B, sh0));
            b1B = fmaxf(0.0f, fmaf(sc1, b1B, sh1));
        }
        if (kk + 4 < in_ch) {                      // next k-slab: one 128B line
            __builtin_prefetch(xA + 4 * S_, 0, 1); // covers both fragments
            __builtin_prefetch(xA + 5 * S_, 0, 1);
        }
        v2f bfA; bfA.x = b0A; bfA.y = b1A;
        v2f bfB; bfB.x = b0B; bfB.y = b1B;
#pragma unroll
        for (int t = 0; t < OUT_TILES; ++t) {
            v2f af;
            af.x = sW[aoff + t * 16 * pitch];
            af.y = sW[aoff + t * 16 * pitch + 1];
            acc[t][0] = __builtin_amdgcn_wmma_f32_16x16x4_f32(
                false, af, false, bfA, (short)0, acc[t][0], false, false);
            acc[t][1] = __builtin_amdgcn_wmma_f32_16x16x4_f32(
                false, af, false, bfB, (short)0, acc[t][1], false, false);
        }
        xA += 4 * S_;
        xB += 4 * S_;
        aoff += 4;
    }

#pragma unroll
    for (int t = 0; t < OUT_TILES; ++t) {
#pragma unroll
        for (int r = 0; r < 8; ++r) {
            const int m = t * 16 + r + 8 * half;
            Y[m * S_ + s0 + col]      = acc[t][0][r];
            Y[m * S_ + s0 + 16 + col] = acc[t][1][r];
        }
    }
}

// ---------------------------------------------------------------------------
// Per-channel train-mode BN stats over S samples; emit fused scale/shift:
//   scale = g * rsqrt(var + eps), shift = b - mean * scale.
__global__ void bn_stats_kernel(const float* __restrict__ Y,
                                const float* __restrict__ g,
                                const float* __restrict__ b,
                                float* __restrict__ scale,
                                float* __restrict__ shift) {
    const int c = blockIdx.x;
    const float* y = Y + c * S_;
    float sum = 0.0f, sq = 0.0f;
    for (int i = threadIdx.x; i < S_; i += 256) {
        float v = y[i];
        sum += v; sq = fmaf(v, v, sq);
    }
    __shared__ float ssum[256], ssq[256];
    ssum[threadIdx.x] = sum; ssq[threadIdx.x] = sq;
    __syncthreads();
    for (int o = 128; o > 0; o >>= 1) {
        if (threadIdx.x < o) {
            ssum[threadIdx.x] += ssum[threadIdx.x + o];
            ssq[threadIdx.x]  += ssq[threadIdx.x + o];
        }
        __syncthreads();
    }
    if (threadIdx.x == 0) {
        float mean = ssum[0] * (1.0f / (float)S_);
        float var  = ssq[0] * (1.0f / (float)S_) - mean * mean;
        float sc   = g[c] * rsqrtf(var + EPS_);
        scale[c] = sc;
        shift[c] = b[c] - mean * sc;
    }
}

// ---------------------------------------------------------------------------
// Output part 1: pass-through pos1.
__global__ void copy_pos1_kernel(const float* __restrict__ pos1, float* __restrict__ out) {
    int t = blockIdx.x * 256 + threadIdx.x;
    out[t] = pos1[t];                              // B*3*N = 98304 elements
}

// Output part 2: feat1_new[b][o][n] = max_k relu(scale3[o]*y3 + shift3[o]).
__global__ void maxpool_kernel(const float* __restrict__ Y3,
                               const float* __restrict__ scale,
                               const float* __restrict__ shift,
                               float* __restrict__ out) {
    int t = blockIdx.x * 256 + threadIdx.x;        // t over B*128*N
    int n = t & (N_ - 1);
    int o = (t >> 12) & 127;
    int b = t >> 19;
    const float* y = Y3 + o * S_ + (b * N_ + n) * K_;
    float sc = scale[o], sh = shift[o];
    float m = 0.0f;                                // ReLU floor
#pragma unroll
    for (int k = 0; k < K_; ++k) m = fmaxf(m, fmaf(sc, y[k], sh));
    out[(B_ * 3 * N_) + (b * 128 + o) * N_ + n] = m;
}

// ---------------------------------------------------------------------------
extern "C" void kernel_launch(void* const* d_in, const int* in_sizes, int n_in,
                              void* d_out, int out_size, void* d_ws, size_t ws_size,
                              hipStream_t stream) {
    const float* pos1    = (const float*)d_in[0];
    const float* pos1_re = (const float*)d_in[1];
    const float* pos2    = (const float*)d_in[2];
    const float* feat1   = (const float*)d_in[3];
    const float* feat2   = (const float*)d_in[4];
    // d_in[5] = ksample (K_ hardcoded)
    const float* W0 = (const float*)d_in[6];
    const float* g0 = (const float*)d_in[7];
    const float* b0 = (const float*)d_in[8];
    const float* W1 = (const float*)d_in[9];
    const float* g1 = (const float*)d_in[10];
    const float* b1 = (const float*)d_in[11];
    const float* W2 = (const float*)d_in[12];
    const float* g2 = (const float*)d_in[13];
    const float* b2 = (const float*)d_in[14];
    float* out = (float*)d_out;

    char*  ws  = (char*)d_ws;
    int*   idx = (int*)(ws + OFF_IDX);
    float* w0p = (float*)(ws + OFF_W0P);
    float* st  = (float*)(ws + OFF_STATS);
    float *sc1 = st,       *sh1 = st + 128;
    float *sc2 = st + 256, *sh2 = st + 384;
    float *sc3 = st + 512, *sh3 = st + 640;
    float* x0  = (float*)(ws + OFF_X0);
    float* y1  = (float*)(ws + OFF_Y1);
    float* y2  = (float*)(ws + OFF_Y2);
    float* y3  = (float*)(ws + OFF_Y3);            // aliases x0 (x0 dead by then)

    pad_w0_kernel<<<(64 * INCH0P + 255) / 256, 256, 0, stream>>>(W0, w0p);
    knn_kernel<<<(B_ * N_) / 256, 256, 0, stream>>>(pos1_re, pos2, idx);
    build_x0_kernel<<<S_ / 256, 256, 0, stream>>>(pos1, pos2, feat1, feat2, idx, x0);

    const int gemm_blocks = S_ / 256;              // 8 waves x 32 samples per block
    gemm_wmma_kernel<4><<<gemm_blocks, 256, 0, stream>>>(w0p, INCH0P, x0, nullptr, nullptr, 0, y1);
    bn_stats_kernel<<<64, 256, 0, stream>>>(y1, g0, b0, sc1, sh1);
    gemm_wmma_kernel<4><<<gemm_blocks, 256, 0, stream>>>(W1, 64, y1, sc1, sh1, 1, y2);
    bn_stats_kernel<<<64, 256, 0, stream>>>(y2, g1, b1, sc2, sh2);
    gemm_wmma_kernel<8><<<gemm_blocks, 256, 0, stream>>>(W2, 64, y2, sc2, sh2, 1, y3);
    bn_stats_kernel<<<128, 256, 0, stream>>>(y3, g2, b2, sc3, sh3);

    copy_pos1_kernel<<<(B_ * 3 * N_) / 256, 256, 0, stream>>>(pos1, out);
    maxpool_kernel<<<(B_ * 128 * N_) / 256, 256, 0, stream>>>(y3, sc3, sh3, out);
}